// BSplineActivation_5016521801728
// MI455X (gfx1250) — compile-verified
//
#include <hip/hip_runtime.h>

#ifndef __has_builtin
#define __has_builtin(x) 0
#endif

#define BLOCK 256

// Per-interval cubic polynomial table: for knot interval j (0..10),
// spline(x) = poly[j].x + poly[j].y*t + poly[j].z*t^2 + poly[j].w*t^3,
// t = local parameter in [0,1). Derived once per block from the 8
// (zero-padded) B-spline coefficients.
__device__ __forceinline__ float bspline_eval(float x, float inv_h, float ofs,
                                              const float4* __restrict__ poly) {
  float xc = fminf(fmaxf(x, -1.0f), 1.0f);
  float u  = fmaf(xc, inv_h, ofs);    // position in knot units: (xc - g0)/h
  float jf = floorf(u);
  float t  = u - jf;
  int   j  = (int)jf;                 // in [0,10] for x in [-1,1]
  float4 p = poly[j];                 // one ds_load_b128, conflict-free
  return fmaf(fmaf(fmaf(p.w, t, p.z), t, p.y), t, p.x);
}

__global__ __launch_bounds__(BLOCK) void bspline_act_kernel(
    const float* __restrict__ x, const float* __restrict__ grid,
    const float* __restrict__ coef, float* __restrict__ out, int n) {
  __shared__ float  craw[8];    // raw coefficients (async-staged)
  __shared__ float4 poly[12];   // 11 intervals used, padded
  const int tid = threadIdx.x;

#if __has_builtin(__builtin_amdgcn_global_load_async_to_lds_b32)
  // CDNA5 async global->LDS path for the coefficient stage.
  if (tid < 8) {
    __builtin_amdgcn_global_load_async_to_lds_b32(
        (__attribute__((address_space(1))) int*)(coef + tid),
        (__attribute__((address_space(3))) int*)(&craw[tid]),
        0, 0);
  }
#if __has_builtin(__builtin_amdgcn_s_wait_asynccnt)
  __builtin_amdgcn_s_wait_asynccnt(0);
#else
  asm volatile("s_wait_asynccnt 0" ::: "memory");
#endif
#else
  if (tid < 8) craw[tid] = coef[tid];
#endif
  __syncthreads();

  // Build the per-interval cubic coefficients. Out-of-range B-spline
  // coefficients are 0 (matches the reference's 8-basis boundary behavior).
  if (tid < 11) {
    const int k = tid - 3;
    float c0 = (k     >= 0 && k     < 8) ? craw[k]     : 0.0f;
    float c1 = (k + 1 >= 0 && k + 1 < 8) ? craw[k + 1] : 0.0f;
    float c2 = (k + 2 >= 0 && k + 2 < 8) ? craw[k + 2] : 0.0f;
    float c3 = (k + 3 >= 0 && k + 3 < 8) ? craw[k + 3] : 0.0f;
    float4 p;
    p.x = (c0 + 4.0f * c1 + c2) * 0.16666667f;           // t^0
    p.y = (c2 - c0) * 0.5f;                               // t^1
    p.z = (c0 - 2.0f * c1 + c2) * 0.5f;                   // t^2
    p.w = fmaf(3.0f, c1 - c2, c3 - c0) * 0.16666667f;     // t^3
    poly[tid] = p;
  }
  __syncthreads();

  // Uniform grid parameters (grid = linspace(-1.2, 1.2, 12)).
  const float g0    = grid[0];
  const float inv_h = 11.0f / (grid[11] - g0);
  const float ofs   = -g0 * inv_h;

  const int     n4 = n >> 2;
  const float4* __restrict__ x4 = (const float4*)x;
  float4*       __restrict__ o4 = (float4*)out;

  const int stride = gridDim.x * BLOCK;
  int i = blockIdx.x * BLOCK + tid;

  // Main vectorized stream: b128 load -> 4x eval -> b128 store.
  for (; i < n4; i += stride) {
    float4 v = x4[i];
    float4 r;
    r.x = bspline_eval(v.x, inv_h, ofs, poly);
    r.y = bspline_eval(v.y, inv_h, ofs, poly);
    r.z = bspline_eval(v.z, inv_h, ofs, poly);
    r.w = bspline_eval(v.w, inv_h, ofs, poly);
    o4[i] = r;
  }

  // Scalar tail (n not divisible by 4).
  for (int k = (n4 << 2) + blockIdx.x * BLOCK + tid; k < n; k += stride)
    out[k] = bspline_eval(x[k], inv_h, ofs, poly);
}

extern "C" void kernel_launch(void* const* d_in, const int* in_sizes, int n_in,
                              void* d_out, int out_size, void* d_ws, size_t ws_size,
                              hipStream_t stream) {
  (void)n_in; (void)out_size; (void)d_ws; (void)ws_size;
  const float* x    = (const float*)d_in[0];  // [2048*4096] f32
  const float* grid = (const float*)d_in[1];  // [12] f32
  const float* coef = (const float*)d_in[2];  // [8] f32
  float*       out  = (float*)d_out;

  const int n  = in_sizes[0];
  const int n4 = n >> 2;
  int blocks = (n4 + BLOCK - 1) / BLOCK;
  if (blocks < 1) blocks = 1;
  if (blocks > 8192) blocks = 8192;   // 8192 x 256 = 2M threads, one float4 each

  bspline_act_kernel<<<blocks, BLOCK, 0, stream>>>(x, grid, coef, out, n);
}